// kNNQuery_48833778155967
// MI455X (gfx1250) — compile-verified
//
#include <hip/hip_runtime.h>

typedef __attribute__((ext_vector_type(2))) float v2f;
typedef __attribute__((ext_vector_type(8))) float v8f;

#define NQv 4096
#define NKv 4096
#define CCv 256
#define KNNv 8
#define FMAXv 3.4028235e38f

// ---------------------------------------------------------------------------
// Kernel 1: kNN via WMMA f32 16x16x4 distance tiles + per-lane top-8.
// A-row (keys)   = (kx, ky, kz, ||k||^2)
// B-col (query)  = (-2qx, -2qy, -2qz, 1)
// D[m,n] = -2 q.k + ||k||^2   (== true dist minus per-query const ||q||^2)
// Wave layout: lane l -> query n = qbase + (l&15); lane l and l^16 split keys.
// ---------------------------------------------------------------------------
__global__ __launch_bounds__(128) void knn_kernel(const float* __restrict__ qc,
                                                  const float* __restrict__ kc,
                                                  int* __restrict__ knn_idx)
{
    __shared__ float4 skey[NKv];            // 64 KB: staged keys + norms
    __shared__ float  s_dist[4][16][16];    // merge scratch
    __shared__ int    s_sidx[4][16][16];

    const int b    = blockIdx.y;
    const int tid  = threadIdx.x;
    const int w    = tid >> 5;
    const int l    = tid & 31;
    const int n16  = l & 15;
    const int half = l >> 4;

    // Stage key coords + norms into LDS once per block.
    const float* kcb = kc + (size_t)b * 3 * NKv;
    for (int i = tid; i < NKv; i += 128) {
        float kx = kcb[i];
        float ky = kcb[NKv + i];
        float kz = kcb[2 * NKv + i];
        skey[i] = make_float4(kx, ky, kz, kx * kx + ky * ky + kz * kz);
    }
    __syncthreads();

    const int qbase = (blockIdx.x * 4 + w) * 16;
    const int n     = qbase + n16;
    const float* qcb = qc + (size_t)b * 3 * NQv;
    const float qx = qcb[n];
    const float qy = qcb[NQv + n];
    const float qz = qcb[2 * NQv + n];

    v2f bmat;
    bmat.x = half ? (-2.0f * qz) : (-2.0f * qx);   // K=2 : K=0
    bmat.y = half ? 1.0f         : (-2.0f * qy);   // K=3 : K=1

    float kd[8];
    int   ki[8];
#pragma unroll
    for (int j = 0; j < 8; ++j) { kd[j] = FMAXv; ki[j] = 0; }
    float mx = FMAXv;

    for (int kt = 0; kt < NKv / 16; ++kt) {
        const int kb = kt * 16;
        float4 kv = skey[kb + n16];
        v2f amat;
        amat.x = half ? kv.z : kv.x;               // K=2 : K=0
        amat.y = half ? kv.w : kv.y;               // K=3 : K=1

        v8f c = {};
        c = __builtin_amdgcn_wmma_f32_16x16x4_f32(
                /*neg_a=*/false, amat, /*neg_b=*/false, bmat,
                /*c_mod=*/(short)0, c, /*reuse_a=*/false, /*reuse_b=*/false);

#pragma unroll
        for (int r = 0; r < 8; ++r) {
            const float d = c[r];
            if (d < mx) {                          // rare after warm-up
                const int cand = kb + half * 8 + r;
                // locate slot holding current max
                int ms = 0; float mv = kd[0];
#pragma unroll
                for (int j = 1; j < 8; ++j) { if (kd[j] > mv) { mv = kd[j]; ms = j; } }
#pragma unroll
                for (int j = 0; j < 8; ++j) { if (ms == j) { kd[j] = d; ki[j] = cand; } }
                // recompute cached max
                float m01 = fmaxf(kd[0], kd[1]), m23 = fmaxf(kd[2], kd[3]);
                float m45 = fmaxf(kd[4], kd[5]), m67 = fmaxf(kd[6], kd[7]);
                mx = fmaxf(fmaxf(m01, m23), fmaxf(m45, m67));
            }
        }
    }

    // Merge the two half-lists (lane l and l^16) per query via LDS selection.
#pragma unroll
    for (int r = 0; r < 8; ++r) {
        s_dist[w][n16][half * 8 + r] = kd[r];
        s_sidx[w][n16][half * 8 + r] = ki[r];
    }
    __syncthreads();

    if (l < 16) {
        int* outp = knn_idx + ((size_t)b * NQv + qbase + l) * KNNv;
        for (int s = 0; s < KNNv; ++s) {
            float best = s_dist[w][l][0]; int bj = 0;
#pragma unroll
            for (int j = 1; j < 16; ++j) {
                float dj = s_dist[w][l][j];
                if (dj < best) { best = dj; bj = j; }
            }
            outp[s] = s_sidx[w][l][bj];
            s_dist[w][l][bj] = FMAXv;
        }
    }
}

// ---------------------------------------------------------------------------
// Kernel 2: gather + subtract + broadcast-concat. Pure bandwidth (~0.8 GiB).
// out[b, c, n, j]      = kf[b, c, idx[b,n,j]] - qf[b, c, n]   (c < C)
// out[b, C + c, n, j]  = qf[b, c, n]
// ---------------------------------------------------------------------------
__global__ __launch_bounds__(256) void gather_kernel(const float* __restrict__ qf,
                                                     const float* __restrict__ kf,
                                                     const int* __restrict__ knn_idx,
                                                     float* __restrict__ out)
{
    const int n = blockIdx.x * 256 + threadIdx.x;
    const int c = blockIdx.y;
    const int b = blockIdx.z;

    const int4 i0 = *(const int4*)(knn_idx + ((size_t)b * NQv + n) * KNNv);
    const int4 i1 = *(const int4*)(knn_idx + ((size_t)b * NQv + n) * KNNv + 4);

    const float q = qf[((size_t)b * CCv + c) * NQv + n];
    const float* krow = kf + ((size_t)b * CCv + c) * NKv;

    float4 o0, o1;
    o0.x = krow[i0.x] - q; o0.y = krow[i0.y] - q;
    o0.z = krow[i0.z] - q; o0.w = krow[i0.w] - q;
    o1.x = krow[i1.x] - q; o1.y = krow[i1.y] - q;
    o1.z = krow[i1.z] - q; o1.w = krow[i1.w] - q;

    const size_t ob0 = (((size_t)b * 2 * CCv + c) * NQv + n) * KNNv;
    *(float4*)(out + ob0)     = o0;
    *(float4*)(out + ob0 + 4) = o1;

    const float4 qv = make_float4(q, q, q, q);
    const size_t ob1 = (((size_t)b * 2 * CCv + CCv + c) * NQv + n) * KNNv;
    *(float4*)(out + ob1)     = qv;
    *(float4*)(out + ob1 + 4) = qv;
}

extern "C" void kernel_launch(void* const* d_in, const int* in_sizes, int n_in,
                              void* d_out, int out_size, void* d_ws, size_t ws_size,
                              hipStream_t stream)
{
    const float* qcoords = (const float*)d_in[0];   // (B, 3, NQ)
    const float* qfeat   = (const float*)d_in[1];   // (B, C, NQ)
    const float* kcoords = (const float*)d_in[2];   // (B, 3, NK)
    const float* kfeat   = (const float*)d_in[3];   // (B, C, NK)
    float* out = (float*)d_out;                     // (B, 2C, NQ, K)

    const int B = in_sizes[0] / (3 * NQv);
    int* knn_idx = (int*)d_ws;                      // B*NQ*K ints = 1 MiB

    // Kernel 1: 4 waves/block, 16 queries/wave -> 64 query-tiles per block-row.
    dim3 g1(NQv / 64, B, 1);
    knn_kernel<<<g1, 128, 0, stream>>>(qcoords, kcoords, knn_idx);

    // Kernel 2: one thread per (b, c, n); 8+8 contiguous f32 out per thread.
    dim3 g2(NQv / 256, CCv, B);
    gather_kernel<<<g2, 256, 0, stream>>>(qfeat, kfeat, knn_idx, out);
}